// RadiusGraph_67997922230586
// MI455X (gfx1250) — compile-verified
//
#include <hip/hip_runtime.h>
#include <math.h>

// CDNA5 wave32 WMMA operand types
typedef float v2f __attribute__((ext_vector_type(2)));
typedef float v8f __attribute__((ext_vector_type(8)));
// async-to-LDS builtin expects 'int __attribute__((vector_size(16))) *'
typedef int   b128_t __attribute__((vector_size(16)));

#define KNN     32
#define CAP     64       // per-query candidate capacity (lambda ~8.8 -> safe)
#define TILE    256      // ref points staged in LDS per iteration
#define QPB     128      // queries per block (8 waves x 16 queries)
#define THREADS 256
#define BSCALE  16.0f    // batch fold scale: mismatch penalty >= 256 >> r^2=4

#if __has_builtin(__builtin_amdgcn_global_load_async_to_lds_b128) && \
    __has_builtin(__builtin_amdgcn_s_wait_asynccnt)
#define USE_ASYNC 1
#else
#define USE_ASYNC 0
#endif

__global__ __launch_bounds__(THREADS) void radius_graph_wmma(
    const float* __restrict__ ref_bxyz,    // [N,4]  (b,x,y,z)
    const float* __restrict__ query_bxyz,  // [M,4]  (b,x,y,z)
    float* __restrict__ out_ref,           // [M*K]  e_ref   as float (-1 pad)
    float* __restrict__ out_query,         // [M*K]  e_query as float (-1 pad)
    float* __restrict__ out_dist,          // [M*K]  e_dist  (0 pad)
    int N)
{
    __shared__ float4 s_ref[2][TILE];     // double-buffered ref tile (b,x,y,z)
    __shared__ float  s_cd[QPB * CAP];    // candidate d^2
    __shared__ int    s_ci[QPB * CAP];    // candidate ref index
    __shared__ int    s_cnt[QPB];         // per-query candidate count

    const int tid   = threadIdx.x;
    const int lane  = tid & 31;
    const int wave  = tid >> 5;
    const int qbase = blockIdx.x * QPB + wave * 16;

    if (tid < QPB) s_cnt[tid] = 0;

    const int n_col = lane & 15;   // column of the 16x16 C tile this lane owns
    const int hi    = lane >> 4;   // lane-half selector per ISA 16x4/4x16 layout

    // ---- A operand: 16x4 f32 query tile, K dims = (x, y, z, BSCALE*b) ----
    // ISA layout: lanes 0-15 hold K=0 (V0) / K=1 (V1); lanes 16-31 hold K=2 / K=3.
    const float4* q4 = (const float4*)query_bxyz;
    const float4* r4 = (const float4*)ref_bxyz;
    float4 qa = q4[qbase + n_col];
    v2f Aop;
    Aop.x = hi ? qa.w : qa.y;              // hi: K2 = z        lo: K0 = x
    Aop.y = hi ? BSCALE * qa.x : qa.z;     // hi: K3 = 16*b     lo: K1 = y

    // Per-lane row constants for the 8 C rows this lane produces: M = i + 8*hi
    float qn2r[8];   // |q|^2 + (16 b)^2
    float qh[8];     // (qn2r - r^2) / 2   (half-space test constant)
#pragma unroll
    for (int i = 0; i < 8; ++i) {
        float4 q = q4[qbase + i + 8 * hi];
        float bn = BSCALE * q.x;
        qn2r[i] = q.y * q.y + q.z * q.z + q.w * q.w + bn * bn;
        qh[i]   = (qn2r[i] - 4.0f) * 0.5f;
    }

    const int nT = N / TILE;
    int buf = 0;

#if USE_ASYNC
    // prefill tile 0 via async global->LDS DMA (ASYNCcnt-tracked)
    __builtin_amdgcn_global_load_async_to_lds_b128(
        (b128_t*)(r4 + tid), (b128_t*)&s_ref[0][tid], 0, 0);
#else
    float4 nxt = r4[tid];
#endif

    // ================= Phase 1: candidate collection =================
    for (int t = 0; t < nT; ++t, buf ^= 1) {
#if USE_ASYNC
        __builtin_amdgcn_s_wait_asynccnt(0);   // my staged tile landed in LDS
        __syncthreads();                        // everyone's tile visible
        if (t + 1 < nT)                         // prefetch next, overlap compute
            __builtin_amdgcn_global_load_async_to_lds_b128(
                (b128_t*)(r4 + (t + 1) * TILE + tid),
                (b128_t*)&s_ref[buf ^ 1][tid], 0, 0);
#else
        __syncthreads();
        s_ref[buf][tid] = nxt;
        __syncthreads();
        if (t + 1 < nT) nxt = r4[(t + 1) * TILE + tid];
#endif
        const float4* rt    = s_ref[buf];
        const int     tbase = t * TILE;

        // unroll 2: two independent subtiles in flight -> both ds_load_b128s
        // issue before either wait; filter ALU of one hides the other's
        // WMMA/LDS latency.
#pragma unroll 2
        for (int sub = 0; sub < TILE; sub += 16) {
            // ---- B operand: 4x16 f32 ref tile (mirror of A layout) ----
            float4 r = rt[sub + n_col];          // this lane's column point
            v2f Bop;
            Bop.x = hi ? r.w : r.y;              // hi: K2 = z      lo: K0 = x
            Bop.y = hi ? BSCALE * r.x : r.z;     // hi: K3 = 16*b   lo: K1 = y

            v8f c = {};
            // c[i] = <(x,y,z,16b)_q , (x,y,z,16b)_r>, row M=i+8*hi, col N=lane&15
            c = __builtin_amdgcn_wmma_f32_16x16x4_f32(
                    false, Aop, false, Bop, (short)0, c, false, false);

            float rbn = BSCALE * r.x;
            float rn2 = r.y * r.y + r.z * r.z + r.w * r.w + rbn * rbn;
            float rh  = 0.5f * rn2;

            // d2 <= r2  <=>  c - qh >= rh  (identical rounding in fast+slow path)
            float tt[8];
#pragma unroll
            for (int i = 0; i < 8; ++i) tt[i] = c[i] - qh[i];
            float mx = fmaxf(fmaxf(fmaxf(tt[0], tt[1]), fmaxf(tt[2], tt[3])),
                             fmaxf(fmaxf(tt[4], tt[5]), fmaxf(tt[6], tt[7])));

            if (mx >= rh) {                      // rare (~1.7% of wave-subtiles)
                int refIdx = tbase + sub + n_col;
#pragma unroll
                for (int i = 0; i < 8; ++i) {
                    if (tt[i] >= rh) {
                        float d2 = fmaf(-2.0f, c[i], qn2r[i] + rn2);
                        int qslot = wave * 16 + i + 8 * hi;
                        int pos = atomicAdd(&s_cnt[qslot], 1);
                        if (pos < CAP) {
                            s_cd[qslot * CAP + pos] = d2;
                            s_ci[qslot * CAP + pos] = refIdx;
                        }
                    }
                }
            }
        }
    }

    __syncthreads();   // all appends visible before selection

    // ================= Phase 2: per-query top-K sort + emit =================
    if (tid < QPB) {
        const int q = blockIdx.x * QPB + tid;
        int cnt = s_cnt[tid];
        if (cnt > CAP) cnt = CAP;
        float* cd = &s_cd[tid * CAP];
        int*   ci = &s_ci[tid * CAP];
        const int sel = cnt < KNN ? cnt : KNN;

        // partial selection sort; tie-break on index -> deterministic output
        for (int k = 0; k < sel; ++k) {
            int   best = k;
            float bd = cd[k];
            int   bi = ci[k];
            for (int j = k + 1; j < cnt; ++j) {
                float dj = cd[j];
                int   ij = ci[j];
                if (dj < bd || (dj == bd && ij < bi)) { best = j; bd = dj; bi = ij; }
            }
            cd[best] = cd[k]; ci[best] = ci[k];
            cd[k] = bd;       ci[k] = bi;

            float d2c = fmaxf(bd, 0.0f);         // clamp (moved out of hot loop)
            out_ref  [q * KNN + k] = (float)bi;
            out_query[q * KNN + k] = (float)q;
            out_dist [q * KNN + k] = sqrtf(d2c);
        }
        for (int k = sel; k < KNN; ++k) {
            out_ref  [q * KNN + k] = -1.0f;
            out_query[q * KNN + k] = -1.0f;
            out_dist [q * KNN + k] = 0.0f;
        }
    }
}

extern "C" void kernel_launch(void* const* d_in, const int* in_sizes, int n_in,
                              void* d_out, int out_size, void* d_ws, size_t ws_size,
                              hipStream_t stream)
{
    const float* ref_bxyz   = (const float*)d_in[0];  // [N,4]
    const float* query_bxyz = (const float*)d_in[1];  // [M,4]
    const int N = in_sizes[0] / 4;
    const int M = in_sizes[1] / 4;

    float* out_ref   = (float*)d_out;                 // e_ref   [M*K]
    float* out_query = out_ref + (size_t)M * KNN;     // e_query [M*K]
    float* out_dist  = out_query + (size_t)M * KNN;   // e_dist  [M*K]

    dim3 grid(M / QPB);
    dim3 block(THREADS);
    radius_graph_wmma<<<grid, block, 0, stream>>>(
        ref_bxyz, query_bxyz, out_ref, out_query, out_dist, N);
}